// SkipGramModel_57166014709963
// MI455X (gfx1250) — compile-verified
//
#include <hip/hip_runtime.h>
#include <math.h>

// CDNA5 / gfx1250: wave32, WMMA f32 16x16x4.
typedef __attribute__((ext_vector_type(2))) float v2f;
typedef __attribute__((ext_vector_type(8))) float v8f;

#define EMBED_DIM 128
#define NEG_K 5
#define WAVES_PER_BLOCK 8

// Numerically stable log_sigmoid(x) = min(x,0) - log1p(exp(-|x|))
__device__ __forceinline__ float log_sigmoid(float x) {
    return fminf(x, 0.0f) - log1pf(expf(-fabsf(x)));
}

// Extract component r (0..7) of an 8-wide fp32 accumulator fragment.
__device__ __forceinline__ float v8f_get(v8f v, int r) {
    switch (r) {
        case 0: return v[0];
        case 1: return v[1];
        case 2: return v[2];
        case 3: return v[3];
        case 4: return v[4];
        case 5: return v[5];
        case 6: return v[6];
        default: return v[7];
    }
}

__global__ void skipgram_zero_out(float* out) { out[0] = 0.0f; }

__global__ __launch_bounds__(32 * WAVES_PER_BLOCK)
void skipgram_wmma_kernel(const int* __restrict__ center_nodes,
                          const int* __restrict__ context_nodes,
                          const int* __restrict__ negative_nodes,
                          const float* __restrict__ u_weight,
                          const float* __restrict__ v_weight,
                          float* __restrict__ out,
                          int batch) {
    const int lane = threadIdx.x & 31;
    const int wave = threadIdx.x >> 5;
    const int tile = blockIdx.x * WAVES_PER_BLOCK + wave;   // 16 batch rows per tile
    const int base = tile * 16;
    const int col  = lane & 15;        // which batch row within the tile (M for A, N for B)
    const int koff = (lane < 16) ? 0 : 2;   // K sub-offset per ISA 16x4 fp32 fragment layout
    const int b    = base + col;

    // Row indices for this lane's matrix column/row.
    const long long ci = (long long)center_nodes[b];
    const long long xi = (long long)context_nodes[b];

    const float* __restrict__ ua = u_weight + (size_t)ci * EMBED_DIM + koff;
    const float* __restrict__ vb[NEG_K + 1];
    vb[0] = v_weight + (size_t)xi * EMBED_DIM + koff;
#pragma unroll
    for (int j = 0; j < NEG_K; ++j) {
        const long long ni = (long long)negative_nodes[b * NEG_K + j];
        vb[j + 1] = v_weight + (size_t)ni * EMBED_DIM + koff;
    }

    // 6 independent accumulator chains: pos + 5 negatives.
    v8f acc[NEG_K + 1] = {};

    // K loop: 128 / 4 = 32 steps of V_WMMA_F32_16X16X4_F32.
#pragma unroll 4
    for (int k = 0; k < EMBED_DIM / 4; ++k) {
        const v2f a = *(const v2f*)(ua + 4 * k);
#pragma unroll
        for (int p = 0; p <= NEG_K; ++p) {
            const v2f bf = *(const v2f*)(vb[p] + 4 * k);
            acc[p] = __builtin_amdgcn_wmma_f32_16x16x4_f32(
                /*neg_a=*/false, a, /*neg_b=*/false, bf,
                /*c_mod=*/(short)0, acc[p],
                /*reuse_a=*/false, /*reuse_b=*/false);
        }
    }

    // Diagonal D[i][i] = dot(u_center[i], v_*[i]).
    // fp32 16x16 C/D layout: VGPR r, lanes 0-15 -> (M=r, N=lane); lanes 16-31 -> (M=r+8, N=lane-16).
    // diag i<8  -> lane i,    component i
    // diag i>=8 -> lane i+16, component lane-24
    const bool valid = (lane < 8) || (lane >= 24);
    const int  r     = (lane < 8) ? lane : (lane - 24);

    float contrib = 0.0f;
    if (valid) {
        float l = log_sigmoid(v8f_get(acc[0], r));
#pragma unroll
        for (int j = 0; j < NEG_K; ++j) {
            l += log_sigmoid(-v8f_get(acc[1 + j], r));
        }
        contrib = -l;
    }

    // Wave32 tree reduction; invalid lanes contribute 0.
#pragma unroll
    for (int off = 16; off > 0; off >>= 1) {
        contrib += __shfl_down(contrib, off, 32);
    }
    if (lane == 0) {
        atomicAdd(out, contrib * (1.0f / (float)batch));
    }
}

extern "C" void kernel_launch(void* const* d_in, const int* in_sizes, int n_in,
                              void* d_out, int out_size, void* d_ws, size_t ws_size,
                              hipStream_t stream) {
    (void)n_in; (void)out_size; (void)d_ws; (void)ws_size;
    const int*   center   = (const int*)d_in[0];
    const int*   context  = (const int*)d_in[1];
    const int*   negative = (const int*)d_in[2];
    const float* u_w      = (const float*)d_in[3];
    const float* v_w      = (const float*)d_in[4];
    float*       out      = (float*)d_out;

    const int batch = in_sizes[0];                 // 65536
    const int tiles = batch / 16;                  // 4096 (batch divisible by 16)
    const int blocks = tiles / WAVES_PER_BLOCK;    // 512

    skipgram_zero_out<<<1, 1, 0, stream>>>(out);
    skipgram_wmma_kernel<<<blocks, 32 * WAVES_PER_BLOCK, 0, stream>>>(
        center, context, negative, u_w, v_w, out, batch);
}